// VecLinear_28982439313683
// MI455X (gfx1250) — compile-verified
//
#include <hip/hip_runtime.h>
#include <cstdint>

typedef __attribute__((ext_vector_type(2))) float v2f;
typedef __attribute__((ext_vector_type(8))) float v8f;
typedef __attribute__((ext_vector_type(4))) unsigned int v4u;
typedef __attribute__((ext_vector_type(4))) int v4i;
typedef __attribute__((ext_vector_type(8))) int v8i;

// ---------------- problem constants ----------------
#define BATCH 8
#define VC 128          // v_in == v_out
#define SC 64           // s_in == s_out
#define NPTS 4096
#define TILE 16         // points per workgroup tile
#define VROWS (VC*3)    // 384 rows of the v tile
#define V_ELEMS ((unsigned long long)BATCH*VC*3*NPTS)   // 12582912

// workspace layout (floats)
#define WS_WSE3   0          // 128*128
#define WS_WCROSS 16384      // 128*128
#define WS_WFC    32768      // 128*256
#define WS_RSV    65536      // 128 rowsum of vsdir_w
#define WS_BCR    65664      // 128 l2n(cross_b)*1e-6
#define WS_BFC    65792      // 128 l2n(crossfc_b)*1e-6
#define WS_BVS    65920      // 128 l2n(vsdir_b)*1e-6
#define WS_FLOATS 66048

// reduction scratch layout (floats, inside LDS "red")
#define R_MVIN 0    // 48 : mean over channels of v_input, per (d,n)
#define R_MO   48   // 48 : mean of v_o (pre-scale)
#define R_MDU  96   // 48 : mean of v_dual
#define R_CN   144  // 16 : ||scale|| over channels
#define R_CN2  160  // 16 : ||rn|| over channels (cev)
#define R_MO2  176  // 48 : mean of v_o (post-scale)
#define R_CNS  224  // 16 : ||s_from_v|| over channels

// LDS float counts
#define L_VIN 0        // 6144  [128][48]
#define L_VO  6144     // 6144
#define L_VB  12288    // 6144
#define L_SIN 18432    // 1024  [64][16]
#define L_SFV 19456    // 2048  [128][16]
#define L_RED 21504    // 256
#define LDS_FLOATS 21760

// ---------------- small helpers ----------------
__device__ __forceinline__ v8f v8f_zero() {
  v8f z;
#pragma unroll
  for (int i = 0; i < 8; ++i) z[i] = 0.f;
  return z;
}

__device__ __forceinline__ unsigned lds_off(const void* p) {
  // generic LDS pointer: low 32 bits == LDS byte offset
  return (unsigned)(unsigned long long)(uintptr_t)p;
}

// D = A(16xK weights) * B(Kx16 activations) + C accumulate, f32 WMMA, K step 4.
// A fragment: lane l -> row m=l&15, k = kh + {0,1}, kh = 2*(l>>4)
// B fragment: lane l -> col n=l&15, k = kh + {0,1}
__device__ __forceinline__ v8f wgemm(const float* __restrict__ W, int ldk, int kofs,
                                     int row0, const float* __restrict__ B,
                                     int rowstride, int rofs, int kcnt, v8f acc) {
  const int lane = threadIdx.x & 31;
  const int m = lane & 15;
  const int kh = (lane >> 4) << 1;
  const float* wr = W + (size_t)(row0 + m) * ldk + kofs + kh;
  const float* br = B + kh * rowstride + rofs + m;
#pragma unroll 4
  for (int k0 = 0; k0 < kcnt; k0 += 4) {
    v2f a, b;
    a.x = wr[0];
    a.y = wr[1];
    b.x = br[0];
    b.y = br[rowstride];
    acc = __builtin_amdgcn_wmma_f32_16x16x4_f32(false, a, false, b, (short)0, acc,
                                                false, false);
    wr += 4;
    br += 4 * rowstride;
  }
  return acc;
}

// scatter a 16x16 f32 D tile to LDS. D layout: vgpr r, lanes0-15 -> M=r, lanes16-31 -> M=r+8
__device__ __forceinline__ void dump8(float* buf, int rowstride, int rofs, int row0,
                                      v8f acc, const float* bias) {
  const int lane = threadIdx.x & 31;
  const int n = lane & 15;
  const int mh = (lane >> 4) << 3;
#pragma unroll
  for (int r = 0; r < 8; ++r) {
    const int o = row0 + mh + r;
    float v = acc[r];
    if (bias) v += bias[o];
    buf[o * rowstride + rofs + n] = v;
  }
}

// ---------------- TDM descriptors (2D tile, 4-byte elements) ----------------
__device__ __forceinline__ void make_desc(v4u& g0, v8i& g1, unsigned lds_byte_off,
                                          unsigned long long gaddr, unsigned tdim0,
                                          unsigned tdim1, unsigned tile0,
                                          unsigned tile1, unsigned long long stride0) {
  g0.x = 1u;                                              // count=1, user descriptor
  g0.y = lds_byte_off;                                    // lds_addr
  g0.z = (unsigned)(gaddr & 0xFFFFFFFFull);               // global_addr lo
  g0.w = (unsigned)((gaddr >> 32) & 0x1FFFFFFull) | (2u << 30);  // hi | type=2
  g1[0] = (int)(2u << 16);                                // data_size=2 (4B)
  g1[1] = (int)((tdim0 & 0xFFFFu) << 16);                 // tensor_dim0[15:0]
  g1[2] = (int)((tdim0 >> 16) | ((tdim1 & 0xFFFFu) << 16));
  g1[3] = (int)((tdim1 >> 16) | (tile0 << 16));           // tile_dim0
  g1[4] = (int)(tile1 & 0xFFFFu);                         // tile_dim1 (tile_dim2=0)
  g1[5] = (int)(stride0 & 0xFFFFFFFFull);                 // tensor_dim0_stride lo
  g1[6] = (int)((stride0 >> 32) & 0xFFFFull);             // stride hi; dim1_stride=0
  g1[7] = 0;
}

__device__ __forceinline__ void tdm_load(v4u g0, v8i g1) {
  v4i z4 = {0, 0, 0, 0};
#if defined(__clang_major__) && __clang_major__ >= 23
  v8i z8 = {0, 0, 0, 0, 0, 0, 0, 0};
  __builtin_amdgcn_tensor_load_to_lds(g0, g1, z4, z4, z8, 0);
#else
  __builtin_amdgcn_tensor_load_to_lds(g0, g1, z4, z4, 0);
#endif
}

__device__ __forceinline__ void tdm_store(v4u g0, v8i g1) {
  v4i z4 = {0, 0, 0, 0};
#if defined(__clang_major__) && __clang_major__ >= 23
  v8i z8 = {0, 0, 0, 0, 0, 0, 0, 0};
  __builtin_amdgcn_tensor_store_from_lds(g0, g1, z4, z4, z8, 0);
#else
  __builtin_amdgcn_tensor_store_from_lds(g0, g1, z4, z4, 0);
#endif
}

// ---------------- pre-pass: SE3 weight completion + bias normalization ----------------
__global__ __launch_bounds__(256) void prep_kernel(
    const float* __restrict__ weight, const float* __restrict__ cross_w,
    const float* __restrict__ cross_b, const float* __restrict__ crossfc_w,
    const float* __restrict__ crossfc_b, const float* __restrict__ vsdir_w,
    const float* __restrict__ vsdir_b, float* __restrict__ ws) {
  __shared__ float nrm[3];
  const int t = threadIdx.x;
  if (t < 3) {
    const float* bp = (t == 0) ? cross_b : ((t == 1) ? crossfc_b : vsdir_b);
    float s = 0.f;
    for (int i = 0; i < VC; ++i) { float v = bp[i]; s += v * v; }
    nrm[t] = fmaxf(sqrtf(s), 1e-12f);
  }
  __syncthreads();
  if (t < VC) {
    const int o = t;
    float s = 0.f;
    for (int c = 0; c < VC - 1; ++c) { float w = weight[o * (VC - 1) + c]; ws[WS_WSE3 + o * VC + c] = w; s += w; }
    ws[WS_WSE3 + o * VC + (VC - 1)] = 1.f - s;
    s = 0.f;
    for (int c = 0; c < VC - 1; ++c) { float w = cross_w[o * (VC - 1) + c]; ws[WS_WCROSS + o * VC + c] = w; s += w; }
    ws[WS_WCROSS + o * VC + (VC - 1)] = 1.f - s;
    s = 0.f;
    for (int c = 0; c < 2 * VC - 1; ++c) { float w = crossfc_w[o * (2 * VC - 1) + c]; ws[WS_WFC + o * 2 * VC + c] = w; s += w; }
    ws[WS_WFC + o * 2 * VC + (2 * VC - 1)] = 1.f - s;
    s = 0.f;
    for (int c = 0; c < VC; ++c) s += vsdir_w[o * VC + c];
    ws[WS_RSV + o] = s;
    ws[WS_BCR + o] = cross_b[o] / nrm[0] * 1e-6f;
    ws[WS_BFC + o] = crossfc_b[o] / nrm[1] * 1e-6f;
    ws[WS_BVS + o] = vsdir_b[o] / nrm[2] * 1e-6f;
  }
}

// ---------------- fused main kernel: one workgroup per (b, 16-point tile) ----------------
__global__ __launch_bounds__(256) void fused_vec_kernel(
    const float* __restrict__ v_in, const float* __restrict__ s_in,
    const float* __restrict__ sv_w, const float* __restrict__ sv_b,
    const float* __restrict__ vsdir_w, const float* __restrict__ vs_w,
    const float* __restrict__ vs_b, const float* __restrict__ ss_w,
    const float* __restrict__ ss_b, const float* __restrict__ ws,
    float* __restrict__ out) {
  extern __shared__ float smem[];
  float* vin = smem + L_VIN;   // [c][d*16+n]  row(c,d) = c*3+d, 16 floats/row
  float* vo  = smem + L_VO;
  float* vb  = smem + L_VB;
  float* sIn = smem + L_SIN;   // [s][n]
  float* sfv = smem + L_SFV;   // [c][n]
  float* red = smem + L_RED;

  const int t = threadIdx.x;
  const int wave = t >> 5;
  const int row0 = wave << 4;
  const int b = blockIdx.x >> 8;
  const int n0 = (blockIdx.x & 255) << 4;

  const float* Wse3   = ws + WS_WSE3;
  const float* Wcross = ws + WS_WCROSS;
  const float* Wfc    = ws + WS_WFC;
  const float* rowsumv = ws + WS_RSV;
  const float* bcr = ws + WS_BCR;
  const float* bfc = ws + WS_BFC;
  const float* bvs = ws + WS_BVS;

  // ---- phase 0: TDM DMA of v tile (384 rows x 64B) and s tile (64 rows x 64B) ----
  if (wave == 0) {
    v4u g0; v8i g1;
    make_desc(g0, g1, lds_off(vin),
              (unsigned long long)(uintptr_t)v_in +
                  (((unsigned long long)b * VROWS * NPTS) + n0) * 4ull,
              NPTS, VROWS, TILE, VROWS, NPTS);
    tdm_load(g0, g1);
    make_desc(g0, g1, lds_off(sIn),
              (unsigned long long)(uintptr_t)s_in +
                  (((unsigned long long)b * SC * NPTS) + n0) * 4ull,
              NPTS, SC, TILE, SC, NPTS);
    tdm_load(g0, g1);
    __builtin_amdgcn_s_wait_tensorcnt(0);
  }
  __syncthreads();

  // ---- phase A: channel mean of v_input; GEMMs v_o, v_dual, scale_pre ----
  if (t < 48) {
    float s = 0.f;
    for (int o = 0; o < VC; ++o) s += vin[o * 48 + t];
    red[R_MVIN + t] = s * (1.f / 128.f);
  }
  v8f accSc = wgemm(sv_w, SC, 0, row0, sIn, 16, 0, SC, v8f_zero());
  v8f acc1[3], accD[3];
#pragma unroll
  for (int d = 0; d < 3; ++d) {
    acc1[d] = wgemm(Wse3, VC, 0, row0, vin, 48, d * 16, VC, v8f_zero());
    accD[d] = wgemm(Wcross, VC, 0, row0, vin, 48, d * 16, VC, v8f_zero());
  }
  dump8(sfv, 16, 0, row0, accSc, sv_b);        // scale_pre
#pragma unroll
  for (int d = 0; d < 3; ++d) {
    dump8(vo, 48, d * 16, row0, acc1[d], nullptr);
    dump8(vb, 48, d * 16, row0, accD[d], bcr); // v_dual (+eps bias)
  }
  __syncthreads();

  // ---- reductions: means of v_o / v_dual, ||scale|| per column ----
  if (t < 48) {
    float s1 = 0.f, s2 = 0.f;
    for (int o = 0; o < VC; ++o) { s1 += vo[o * 48 + t]; s2 += vb[o * 48 + t]; }
    red[R_MO + t] = s1 * (1.f / 128.f);
    red[R_MDU + t] = s2 * (1.f / 128.f);
  } else if (t >= 64 && t < 80) {
    const int n = t - 64;
    float s = 0.f;
    for (int o = 0; o < VC; ++o) { float v = sfv[o * 16 + n]; s += v * v; }
    red[R_CN + n] = fmaxf(sqrtf(s), 1e-12f);
  }
  __syncthreads();

  // ---- pass1: apply scale to v_o; center v_dual -> unit dirs (vb) + norms (sfv) ----
  for (int i = t; i < VC * TILE; i += 256) {
    const int o = i >> 4, n = i & 15;
    const float sc = sfv[i] / red[R_CN + n];
    float rr[3];
#pragma unroll
    for (int d = 0; d < 3; ++d) {
      const int idx = o * 48 + d * 16 + n;
      const float mo = red[R_MO + d * 16 + n];
      vo[idx] = (vo[idx] - mo) * sc + mo;
      rr[d] = vb[idx] - red[R_MDU + d * 16 + n];
    }
    const float rn = sqrtf(rr[0] * rr[0] + rr[1] * rr[1] + rr[2] * rr[2]);
    const float inv = 1.f / fmaxf(rn, 1e-12f);
#pragma unroll
    for (int d = 0; d < 3; ++d) vb[o * 48 + d * 16 + n] = rr[d] * inv;
    sfv[i] = rn;
  }
  __syncthreads();

  // ---- reductions: mean of scaled v_o, ||rn|| per column ----
  if (t < 48) {
    float s = 0.f;
    for (int o = 0; o < VC; ++o) s += vo[o * 48 + t];
    red[R_MO2 + t] = s * (1.f / 128.f);
  } else if (t >= 64 && t < 80) {
    const int n = t - 64;
    float s = 0.f;
    for (int o = 0; o < VC; ++o) { float v = sfv[o * 16 + n]; s += v * v; }
    red[R_CN2 + n] = fmaxf(sqrtf(s), 1e-12f);
  }
  __syncthreads();

  // ---- pass2: v_cross = cross(cev_norm(v_dual - mdual), v_o - mo2) + v_o  (into vb) ----
  for (int i = t; i < VC * TILE; i += 256) {
    const int o = i >> 4, n = i & 15;
    const float f = sfv[i] / red[R_CN2 + n];
    const int i0 = o * 48 + n;
    const float c0 = vb[i0] * f, c1 = vb[i0 + 16] * f, c2 = vb[i0 + 32] * f;
    const float v0 = vo[i0], v1 = vo[i0 + 16], v2 = vo[i0 + 32];
    const float w0 = v0 - red[R_MO2 + n];
    const float w1 = v1 - red[R_MO2 + 16 + n];
    const float w2 = v2 - red[R_MO2 + 32 + n];
    vb[i0]      = c1 * w2 - c2 * w1 + v0;
    vb[i0 + 16] = c2 * w0 - c0 * w2 + v1;
    vb[i0 + 32] = c0 * w1 - c1 * w0 + v2;
  }
  __syncthreads();

  // ---- GEMM crossfc (K=256: vb then vo) and GEMM vsdir on v_input ----
  v8f accF[3], accS[3];
#pragma unroll
  for (int d = 0; d < 3; ++d) {
    v8f a = wgemm(Wfc, 2 * VC, 0, row0, vb, 48, d * 16, VC, v8f_zero());
    accF[d] = wgemm(Wfc, 2 * VC, VC, row0, vo, 48, d * 16, VC, a);
    accS[d] = wgemm(vsdir_w, VC, 0, row0, vin, 48, d * 16, VC, v8f_zero());
  }
  __syncthreads();  // all waves done reading vb/vo

  // final v_o -> vo ; vsdir output (mean-corrected + eps bias) -> vb
#pragma unroll
  for (int d = 0; d < 3; ++d) dump8(vo, 48, d * 16, row0, accF[d], bfc);
  {
    const int lane = t & 31;
    const int n = lane & 15;
    const int mh = (lane >> 4) << 3;
#pragma unroll
    for (int d = 0; d < 3; ++d)
#pragma unroll
      for (int r = 0; r < 8; ++r) {
        const int o = row0 + mh + r;
        vb[o * 48 + d * 16 + n] =
            accS[d][r] - rowsumv[o] * red[R_MVIN + d * 16 + n] + bvs[o];
      }
  }
  __syncthreads();

  // ---- pass3: dual_dir = l2n(vb, dim); s_pre = <v_sR, dual_dir> -> sfv ----
  for (int i = t; i < VC * TILE; i += 256) {
    const int o = i >> 4, n = i & 15;
    const int i0 = o * 48 + n;
    const float r0 = vb[i0], r1 = vb[i0 + 16], r2 = vb[i0 + 32];
    const float inv = 1.f / fmaxf(sqrtf(r0 * r0 + r1 * r1 + r2 * r2), 1e-12f);
    float s = (vin[i0] - red[R_MVIN + n]) * r0;
    s += (vin[i0 + 16] - red[R_MVIN + 16 + n]) * r1;
    s += (vin[i0 + 32] - red[R_MVIN + 32 + n]) * r2;
    sfv[i] = s * inv;
  }
  __syncthreads();
  if (t < 16) {
    float s = 0.f;
    for (int o = 0; o < VC; ++o) { float v = sfv[o * 16 + t]; s += v * v; }
    red[R_CNS + t] = fmaxf(sqrtf(s), 1e-12f);
  }
  __syncthreads();
  for (int i = t; i < VC * TILE; i += 256) sfv[i] /= red[R_CNS + (i & 15)];
  __syncthreads();

  // ---- scalar-output GEMMs: vs_w*s_from_v (waves 0-3), ss_w*s_in (waves 4-7) ----
  if (wave < 4) {
    v8f a = wgemm(vs_w, VC, 0, row0, sfv, 16, 0, VC, v8f_zero());
    dump8(vb, 16, 0, row0, a, vs_b);
  } else {
    const int r0s = (wave - 4) << 4;
    v8f a = wgemm(ss_w, SC, 0, r0s, sIn, 16, 0, SC, v8f_zero());
    dump8(vb + 1024, 16, 0, r0s, a, ss_b);
  }
  __syncthreads();
  for (int i = t; i < SC * TILE; i += 256) vb[2048 + i] = vb[i] + vb[1024 + i];
  __syncthreads();

  // ---- phase Z: TDM store of v_o (384x16) and s_out (64x16) ----
  if (wave == 0) {
    v4u g0; v8i g1;
    make_desc(g0, g1, lds_off(vo),
              (unsigned long long)(uintptr_t)out +
                  (((unsigned long long)b * VROWS * NPTS) + n0) * 4ull,
              NPTS, VROWS, TILE, VROWS, NPTS);
    tdm_store(g0, g1);
    make_desc(g0, g1, lds_off(vb + 2048),
              (unsigned long long)(uintptr_t)out +
                  (V_ELEMS + (unsigned long long)b * SC * NPTS + n0) * 4ull,
              NPTS, SC, TILE, SC, NPTS);
    tdm_store(g0, g1);
    __builtin_amdgcn_s_wait_tensorcnt(0);
  }
}

extern "C" void kernel_launch(void* const* d_in, const int* in_sizes, int n_in,
                              void* d_out, int out_size, void* d_ws, size_t ws_size,
                              hipStream_t stream) {
  const float* v_in      = (const float*)d_in[0];
  const float* s_in      = (const float*)d_in[1];
  const float* weight    = (const float*)d_in[2];
  const float* sv_w      = (const float*)d_in[3];
  const float* sv_b      = (const float*)d_in[4];
  const float* cross_w   = (const float*)d_in[5];
  const float* cross_b   = (const float*)d_in[6];
  const float* crossfc_w = (const float*)d_in[7];
  const float* crossfc_b = (const float*)d_in[8];
  const float* vsdir_w   = (const float*)d_in[9];
  const float* vsdir_b   = (const float*)d_in[10];
  const float* vs_w      = (const float*)d_in[11];
  const float* vs_b      = (const float*)d_in[12];
  const float* ss_w      = (const float*)d_in[13];
  const float* ss_b      = (const float*)d_in[14];
  float* ws  = (float*)d_ws;
  float* out = (float*)d_out;

  prep_kernel<<<1, 256, 0, stream>>>(weight, cross_w, cross_b, crossfc_w,
                                     crossfc_b, vsdir_w, vsdir_b, ws);
  const int nblocks = BATCH * (NPTS / TILE);  // 2048
  fused_vec_kernel<<<nblocks, 256, LDS_FLOATS * sizeof(float), stream>>>(
      v_in, s_in, sv_w, sv_b, vsdir_w, vs_w, vs_b, ss_w, ss_b, ws, out);
}